// CoscamLoss_57921928954243
// MI455X (gfx1250) — compile-verified
//
#include <hip/hip_runtime.h>
#include <hip/hip_bf16.h>

// ---------------------------------------------------------------------------
// CoscamLoss for MI455X (gfx1250, wave32).
// Memory-bound streaming kernel: 512 MB read once -> ~22 us floor @ 23.3 TB/s.
// Kernel 1: per-row online logsumexp; double-buffered GLOBAL_LOAD_ASYNC_TO_LDS
//           _B128 pipeline with TH=NT (pure streaming, no reuse), waits via
//           s_wait_asynccnt.
// Kernel 2: mean of row losses via V_WMMA_F32_16X16X4_F32 ones-reduction.
// ---------------------------------------------------------------------------

#define B_ROWS 4096
#define C_COLS 16384
#define SCALE_F 16.0f
#define MARGIN_F 0.1f
#define NEG_INF_F (-10000.0f)
#define HARD_SCALE_F 1.012f
#define HARD_SHIFT_F 0.012f
#define L2E 1.44269504088896340736f   // log2(e)
#define LN2 0.6931471805599453f

#define TILE 2048        // floats per array per stage (8 KB): 2x b128 / thread
#define NTHREADS 256     // 8 wave32 waves
#define CPOL_NT 1        // CPol TH[2:0]=1 -> non-temporal load (streaming)

typedef __attribute__((ext_vector_type(2))) float v2f;
typedef __attribute__((ext_vector_type(8))) float v8f;
// Matches the async builtin's parameter type (round-1 diagnostic):
//   "__attribute__((__vector_size__(4 * sizeof(int)))) int __device__ *"
typedef int vs4i __attribute__((vector_size(16)));
typedef __attribute__((address_space(1))) vs4i* as1_v4i_ptr;
typedef __attribute__((address_space(3))) vs4i* as3_v4i_ptr;

// ---- gfx1250 feature probes (guarded, with compiling fallbacks) ------------
#if __has_builtin(__builtin_amdgcn_global_load_async_to_lds_b128)
#define HAVE_ASYNC_LDS 1
#else
#define HAVE_ASYNC_LDS 0
#endif

#if __has_builtin(__builtin_amdgcn_s_wait_asynccnt)
#define WAIT_ASYNC(n) __builtin_amdgcn_s_wait_asynccnt(n)
#else
#define WAIT_ASYNC(n) asm volatile("s_wait_asynccnt %0" ::"i"(n))
#endif

#if __has_builtin(__builtin_amdgcn_exp2f)
#define EXP2F(x) __builtin_amdgcn_exp2f(x)
#else
#define EXP2F(x) exp2f(x)
#endif

#if __has_builtin(__builtin_amdgcn_logf)
#define LOG2F(x) __builtin_amdgcn_logf(x)
#else
#define LOG2F(x) log2f(x)
#endif

// Async copy of 16 bytes per lane: global -> LDS, tracked by ASYNCcnt.
__device__ __forceinline__ void async_b128(const float* gsrc, float* ldst) {
#if HAVE_ASYNC_LDS
  __builtin_amdgcn_global_load_async_to_lds_b128(
      (as1_v4i_ptr)(void*)const_cast<float*>(gsrc),
      (as3_v4i_ptr)(void*)ldst,
      /*offset=*/0, /*cpol=*/CPOL_NT);
#else
  // Fallback: synchronous vector load + LDS store (still b128 traffic).
  *(float4*)ldst = *(const float4*)gsrc;
#endif
}

// ---------------------------------------------------------------------------
// Kernel 1: one block per row; online logsumexp of the modified logits.
// ---------------------------------------------------------------------------
__global__ __launch_bounds__(NTHREADS) void coscam_row_kernel(
    const float* __restrict__ inputs,     // (B, C)
    const int* __restrict__ targets,      // (B,)
    const float* __restrict__ pos_mask,   // (B, C) 0/1 floats
    float* __restrict__ row_loss) {       // (B,)
  const int row = blockIdx.x;
  const int tid = threadIdx.x;

  __shared__ float lds_in[2][TILE];   // 16 KB
  __shared__ float lds_pm[2][TILE];   // 16 KB
  __shared__ float red_m[NTHREADS];   // 1 KB
  __shared__ float red_s[NTHREADS];   // 1 KB
  __shared__ float s_gt;
  __shared__ int s_tgt;

  const long rowoff = (long)row * C_COLS;
  const float* gin = inputs + rowoff;
  const float* gpm = pos_mask + rowoff;

  if (tid == 0) {
    const int tgt = targets[row];
    s_tgt = tgt;
    s_gt = gin[tgt];  // pre-modification target logit
  }
  __syncthreads();
  const int tgt = s_tgt;
  const float gt = s_gt;
  const float o_tgt = SCALE_F * (gt - MARGIN_F);  // output at target position

  const int NT = C_COLS / TILE;  // 8 stages

  // Prologue: stage 0 into buffer 0. Each thread moves two float4 per array
  // (lanes cover [tid*4, tid*4+3] and [1024 + tid*4 ...]).
  async_b128(gin + (size_t)tid * 4, &lds_in[0][tid * 4]);
  async_b128(gin + 1024 + (size_t)tid * 4, &lds_in[0][1024 + tid * 4]);
  async_b128(gpm + (size_t)tid * 4, &lds_pm[0][tid * 4]);
  async_b128(gpm + 1024 + (size_t)tid * 4, &lds_pm[0][1024 + tid * 4]);

  float m = -3.0e38f;  // running max of outputs o
  float s = 0.0f;      // running sum of exp(o - m)

  for (int t = 0; t < NT; ++t) {
    const int buf = t & 1;
    if (t + 1 < NT) {
      const size_t go = (size_t)(t + 1) * TILE + (size_t)tid * 4;
      async_b128(gin + go, &lds_in[buf ^ 1][tid * 4]);
      async_b128(gin + go + 1024, &lds_in[buf ^ 1][1024 + tid * 4]);
      async_b128(gpm + go, &lds_pm[buf ^ 1][tid * 4]);
      async_b128(gpm + go + 1024, &lds_pm[buf ^ 1][1024 + tid * 4]);
      WAIT_ASYNC(4);  // retire stage t (4 older instrs); keep t+1 in flight
    } else {
      WAIT_ASYNC(0);
    }
    __syncthreads();  // all waves' tiles visible

#pragma unroll
    for (int h = 0; h < 2; ++h) {
      const int base = h * 1024 + tid * 4;
      const float4 vi = *(const float4*)&lds_in[buf][base];
      const float4 vp = *(const float4*)&lds_pm[buf][base];
      const int cbase = t * TILE + base;

#pragma unroll
      for (int j = 0; j < 4; ++j) {
        const float in = (&vi.x)[j];
        const float pm = (&vp.x)[j];
        // cam = pm*in + (1-pm)*NEG_INF   (pm is exactly 0.0 or 1.0)
        const float cam = pm * in + (1.0f - pm) * NEG_INF_F;
        const bool hard = cam >= gt;
        const float x = hard ? (HARD_SCALE_F * in + HARD_SHIFT_F) : in;
        float o = SCALE_F * x;
        if (cbase + j == tgt) o = o_tgt;  // scatter-restore + margin at target
        // Online logsumexp update (o can reach ~|100|: exp must be shifted).
        if (o > m) {
          s = s * EXP2F((m - o) * L2E) + 1.0f;
          m = o;
        } else {
          s += EXP2F((o - m) * L2E);
        }
      }
    }
    __syncthreads();  // tile fully consumed before buffer reuse (t+2 overwrite)
  }

  // Block tree-combine of (m, s) pairs.
  red_m[tid] = m;
  red_s[tid] = s;
  __syncthreads();
  for (int off = NTHREADS / 2; off > 0; off >>= 1) {
    if (tid < off) {
      const float m1 = red_m[tid], s1 = red_s[tid];
      const float m2 = red_m[tid + off], s2 = red_s[tid + off];
      const float M = fmaxf(m1, m2);
      red_s[tid] = s1 * EXP2F((m1 - M) * L2E) + s2 * EXP2F((m2 - M) * L2E);
      red_m[tid] = M;
    }
    __syncthreads();
  }

  if (tid == 0) {
    const float lse = red_m[0] + LOG2F(red_s[0]) * LN2;  // ln-sum-exp
    row_loss[row] = lse - o_tgt;                          // -log p[target]
  }
}

// ---------------------------------------------------------------------------
// Kernel 2: mean of B row losses with a WMMA ones-matrix reduction.
// D = ones(16x4) x B(4x16): every row of D holds the column sums of B, so
// summing D[vgpr0] over all 32 lanes counts every B element exactly twice
// (lanes 0-15 carry M=0, lanes 16-31 carry M=8) -- layout-agnostic exact sum.
// ---------------------------------------------------------------------------
__global__ __launch_bounds__(32) void coscam_mean_kernel(
    const float* __restrict__ row_loss, float* __restrict__ out, int n) {
  const int lane = threadIdx.x;  // single wave32
  float part = 0.0f;
  for (int i = lane; i < n; i += 32) part += row_loss[i];

#if __has_builtin(__builtin_amdgcn_wmma_f32_16x16x4_f32)
  v2f a;
  a.x = 1.0f;
  a.y = 1.0f;             // A = ones(16x4)
  v2f b;
  b.x = part;
  b.y = 0.0f;             // each partial occupies exactly one B(4x16) slot
  v8f c = {};
  c = __builtin_amdgcn_wmma_f32_16x16x4_f32(
      /*neg_a=*/false, a, /*neg_b=*/false, b,
      /*c_mod=*/(short)0, c, /*reuse_a=*/false, /*reuse_b=*/false);
  float d0 = c[0];  // lane L: D[M=0 or 8][N=L%16] = colsum(B, N)
  for (int off = 16; off > 0; off >>= 1) d0 += __shfl_xor(d0, off, 32);
  if (lane == 0) out[0] = d0 * 0.5f / (float)n;  // /2: rows 0 and 8 both summed
#else
  for (int off = 16; off > 0; off >>= 1) part += __shfl_xor(part, off, 32);
  if (lane == 0) out[0] = part / (float)n;
#endif
}

// ---------------------------------------------------------------------------
// Harness entry. Inputs: [0]=inputs f32 (B*C), [1]=targets int (B),
// [2]=mask (unused), [3]=pos_cam_mask f32 (B*C). Output: scalar f32 loss.
// ---------------------------------------------------------------------------
extern "C" void kernel_launch(void* const* d_in, const int* in_sizes, int n_in,
                              void* d_out, int out_size, void* d_ws,
                              size_t ws_size, hipStream_t stream) {
  (void)n_in;
  (void)out_size;
  (void)ws_size;
  const float* inputs = (const float*)d_in[0];
  const int* targets = (const int*)d_in[1];
  const float* pos_mask = (const float*)d_in[3];
  float* out = (float*)d_out;
  float* row_loss = (float*)d_ws;  // B floats of scratch

  const int b = in_sizes[1];  // 4096
  coscam_row_kernel<<<dim3(b), dim3(NTHREADS), 0, stream>>>(inputs, targets,
                                                            pos_mask, row_loss);
  coscam_mean_kernel<<<dim3(1), dim3(32), 0, stream>>>(row_loss, out, b);
}